// OPnnLayer_71545565217479
// MI455X (gfx1250) — compile-verified
//
#include <hip/hip_runtime.h>
#include <hip/hip_bf16.h>

// OPnnLayer: out[b, :, :] = s_b (D x 1) * s_b^T (1 x D),  s_b = sum_f in[f,b,:]
// F=32, B=8192, D=64.  Memory-bound: 64 MB in + 128 MB out -> ~8.2 us floor
// at 23.3 TB/s.  Fused single pass; outer product done on the WMMA pipe as a
// K=1 GEMM via V_WMMA_F32_16X16X4_F32 (16x16 tiles, 4x4 tiles per sample).

typedef float v2f __attribute__((ext_vector_type(2)));
typedef float v8f __attribute__((ext_vector_type(8)));

#define NFIELDS 32
#define BATCH   8192
#define DIM     64
#define WAVES_PER_BLOCK 8

__global__ __launch_bounds__(WAVES_PER_BLOCK * 32)
void opnn_outer_wmma(const float* __restrict__ in, float* __restrict__ out) {
    __shared__ float sbuf[WAVES_PER_BLOCK][DIM];

    const int lane = threadIdx.x & 31;
    const int wave = threadIdx.x >> 5;
    const int b    = blockIdx.x * WAVES_PER_BLOCK + wave;   // grid sized exactly

    // ---- Field reduction: s[d] = sum_f in[f*B*D + b*D + d] ----
    // lane handles d = 2*lane, 2*lane+1 -> 256 B contiguous per wave per field.
    const float* p = in + (size_t)b * DIM + (size_t)(lane * 2);
    float ax = 0.0f, ay = 0.0f;
    #pragma unroll
    for (int f = 0; f < NFIELDS; ++f) {
        const float2 v = *(const float2*)(p + (size_t)f * (BATCH * DIM));
        ax += v.x;
        ay += v.y;
    }
    sbuf[wave][lane * 2 + 0] = ax;
    sbuf[wave][lane * 2 + 1] = ay;
    __syncthreads();   // publish s to LDS before lane-redistributed reads

    // ---- Outer product: 4x4 grid of 16x16 WMMA tiles, K=1 ----
    // A (16x4 f32): lanes 0-15 VGPR0 = K=0 column chunk; all other slots 0.
    // B (4x16 f32): lanes 0-15 VGPR0 = K=0 row chunk;   all other slots 0.
    const int  l15  = lane & 15;
    const int  half = lane >> 4;          // 0: rows m..m ; 1: rows m+8
    float* outb = out + (size_t)b * (DIM * DIM);

    #pragma unroll
    for (int tm = 0; tm < 4; ++tm) {
        const float acol = sbuf[wave][tm * 16 + l15];
        v2f A;
        A.x = half ? 0.0f : acol;         // K=0 (lanes 0-15); K=2 slot zeroed
        A.y = 0.0f;                       // K=1 / K=3 slots

        #pragma unroll
        for (int tn = 0; tn < 4; ++tn) {
            const float brow = sbuf[wave][tn * 16 + l15];
            v2f Bv;
            Bv.x = half ? 0.0f : brow;    // K=0 row, N = lane
            Bv.y = 0.0f;

            v8f c = {};
            // (neg_a, A, neg_b, B, c_mod, C, reuse_a, reuse_b)
            c = __builtin_amdgcn_wmma_f32_16x16x4_f32(
                    false, A, false, Bv, (short)0, c, false, false);

            // C/D layout: VGPR r -> row (tm*16 + r + 8*half), col (tn*16 + l15)
            #pragma unroll
            for (int r = 0; r < 8; ++r) {
                const int m = tm * 16 + r + 8 * half;
                const int n = tn * 16 + l15;
                outb[m * DIM + n] = c[r];
            }
        }
    }
}

extern "C" void kernel_launch(void* const* d_in, const int* in_sizes, int n_in,
                              void* d_out, int out_size, void* d_ws, size_t ws_size,
                              hipStream_t stream) {
    (void)in_sizes; (void)n_in; (void)d_ws; (void)ws_size; (void)out_size;
    const float* in  = (const float*)d_in[0];
    float*       out = (float*)d_out;
    const int blocks = BATCH / WAVES_PER_BLOCK;   // 1024
    opnn_outer_wmma<<<blocks, WAVES_PER_BLOCK * 32, 0, stream>>>(in, out);
}